// SmplxLite_6880537608611
// MI455X (gfx1250) — compile-verified
//
#include <hip/hip_runtime.h>
#include <hip/hip_bf16.h>
#include <math.h>

#define V_    10475
#define J_    55
#define NB_   10
#define PF_   486
#define KP_   488          // PF padded to multiple of 4
#define NCOLS (V_ * 3)     // 31425
#define FR_   512          // B*L
#define BN_   128
#define BM_   32
#define AST_  488          // LDS row stride (TDM writes rows contiguously)

typedef __attribute__((ext_vector_type(2))) float v2f;
typedef __attribute__((ext_vector_type(8))) float v8f;
typedef __attribute__((ext_vector_type(4))) unsigned int u32x4;
typedef __attribute__((ext_vector_type(4))) int i32x4;
typedef __attribute__((ext_vector_type(8))) int i32x8;

__constant__ int c_parents[J_] = {
    -1, 0, 0, 0, 1, 2, 3, 4, 5, 6, 7, 8, 9, 9, 9, 12, 13, 14, 16, 17, 18, 19,
    15, 15, 15,
    20, 25, 26, 20, 28, 29, 20, 31, 32, 20, 34, 35, 20, 37, 38,
    21, 40, 41, 21, 43, 44, 21, 46, 47, 21, 49, 50, 21, 52, 53};

// ---------------------------------------------------------------------------
// Kernel 0: transpose lbs_weights (V,55) -> (55,V) for coalesced skinning reads
// ---------------------------------------------------------------------------
__global__ __launch_bounds__(256) void k_transpose_w(const float* __restrict__ w,
                                                     float* __restrict__ wT) {
    int i = blockIdx.x * 256 + threadIdx.x;
    if (i < V_ * J_) {
        int j = i / V_;
        int v = i - j * V_;
        wT[i] = w[(size_t)v * J_ + j];
    }
}

// ---------------------------------------------------------------------------
// Kernel 1: per-frame pose math. 1 block (64 thr) per frame.
// Produces: pf_buf[f][488] (pose feature, zero padded), A_buf[f][55][16].
// ---------------------------------------------------------------------------
__global__ __launch_bounds__(64) void k_pose(const float* __restrict__ body_pose,
                                             const float* __restrict__ betas,
                                             const float* __restrict__ global_orient,
                                             const float* __restrict__ J_template,
                                             const float* __restrict__ J_shapedirs,
                                             const float* __restrict__ odp,
                                             float* __restrict__ pf_buf,
                                             float* __restrict__ A_buf) {
    const int f = blockIdx.x;
    const int t = threadIdx.x;
    __shared__ float rot[J_ * 9];
    __shared__ float jt[J_ * 3];
    __shared__ float loc[J_ * 16];
    __shared__ float chn[J_ * 16];

    if (t < J_) {
        // gather axis-angle for joint t from concat(global_orient, body_pose, odp)
        float ax, ay, az;
        if (t == 0) {
            ax = global_orient[f * 3 + 0];
            ay = global_orient[f * 3 + 1];
            az = global_orient[f * 3 + 2];
        } else if (t < 22) {
            const float* p = body_pose + (size_t)f * 63 + (t - 1) * 3;
            ax = p[0]; ay = p[1]; az = p[2];
        } else {
            const float* p = odp + (t - 22) * 3;
            ax = p[0]; ay = p[1]; az = p[2];
        }
        float ang = sqrtf(ax * ax + ay * ay + az * az + 1e-12f);
        float inv = 1.0f / ang;
        float x = ax * inv, y = ay * inv, z = az * inv;
        float s = sinf(ang), c = cosf(ang), tc = 1.0f - c;
        float* R = &rot[t * 9];
        R[0] = c + tc * x * x;  R[1] = tc * x * y - s * z; R[2] = tc * x * z + s * y;
        R[3] = tc * x * y + s * z; R[4] = c + tc * y * y;  R[5] = tc * y * z - s * x;
        R[6] = tc * x * z - s * y; R[7] = tc * y * z + s * x; R[8] = c + tc * z * z;
        // joints = J_template + J_shapedirs . betas
        for (int cd = 0; cd < 3; ++cd) {
            float v = J_template[t * 3 + cd];
            #pragma unroll
            for (int k = 0; k < NB_; ++k)
                v += betas[(size_t)f * NB_ + k] * J_shapedirs[(t * 3 + cd) * NB_ + k];
            jt[t * 3 + cd] = v;
        }
    }
    __syncthreads();

    if (t < J_) {
        int p = c_parents[t];
        float rx, ry, rz;
        if (t == 0) { rx = jt[0]; ry = jt[1]; rz = jt[2]; }
        else {
            rx = jt[t * 3 + 0] - jt[p * 3 + 0];
            ry = jt[t * 3 + 1] - jt[p * 3 + 1];
            rz = jt[t * 3 + 2] - jt[p * 3 + 2];
        }
        float* L = &loc[t * 16];
        const float* R = &rot[t * 9];
        L[0] = R[0]; L[1] = R[1]; L[2]  = R[2]; L[3]  = rx;
        L[4] = R[3]; L[5] = R[4]; L[6]  = R[5]; L[7]  = ry;
        L[8] = R[6]; L[9] = R[7]; L[10] = R[8]; L[11] = rz;
        L[12] = 0.f; L[13] = 0.f; L[14] = 0.f; L[15] = 1.f;
        if (t >= 1) {  // pose feature = R - I, row major
            float* pf = pf_buf + (size_t)f * KP_ + (t - 1) * 9;
            pf[0] = R[0] - 1.f; pf[1] = R[1];       pf[2] = R[2];
            pf[3] = R[3];       pf[4] = R[4] - 1.f; pf[5] = R[5];
            pf[6] = R[6];       pf[7] = R[7];       pf[8] = R[8] - 1.f;
        }
    }
    if (t < 2) pf_buf[(size_t)f * KP_ + PF_ + t] = 0.f;  // K padding
    __syncthreads();

    if (t < 16) chn[t] = loc[t];  // chain root
    __syncthreads();
    for (int j = 1; j < J_; ++j) {  // fixed kinematic tree, 16 threads per 4x4
        if (t < 16) {
            int p = c_parents[j];
            int r = t >> 2, cc = t & 3;
            float s = 0.f;
            #pragma unroll
            for (int kk = 0; kk < 4; ++kk)
                s += chn[p * 16 + r * 4 + kk] * loc[j * 16 + kk * 4 + cc];
            chn[j * 16 + t] = s;
        }
        __syncthreads();
    }

    if (t < J_) {  // subtract T[:3,:3] @ joints from translation column
        float* A = A_buf + (size_t)f * (J_ * 16) + t * 16;
        #pragma unroll
        for (int r = 0; r < 3; ++r) {
            float corr = chn[t * 16 + r * 4 + 0] * jt[t * 3 + 0] +
                         chn[t * 16 + r * 4 + 1] * jt[t * 3 + 1] +
                         chn[t * 16 + r * 4 + 2] * jt[t * 3 + 2];
            A[r * 4 + 0] = chn[t * 16 + r * 4 + 0];
            A[r * 4 + 1] = chn[t * 16 + r * 4 + 1];
            A[r * 4 + 2] = chn[t * 16 + r * 4 + 2];
            A[r * 4 + 3] = chn[t * 16 + r * 4 + 3] - corr;
        }
        A[12] = 0.f; A[13] = 0.f; A[14] = 0.f; A[15] = 1.f;
    }
}

// ---------------------------------------------------------------------------
// Kernel 2: v_posed = pose_feature @ posedirs + betas @ shapedirs + v_template
// WMMA f32 16x16x4; A tile staged via Tensor Data Mover (one 4x488 slice/wave).
// ---------------------------------------------------------------------------
__global__ __launch_bounds__(256) void k_gemm_pose(const float* __restrict__ pf_buf,
                                                   const float* __restrict__ posedirs,
                                                   const float* __restrict__ shapedirs,
                                                   const float* __restrict__ v_template,
                                                   const float* __restrict__ betas,
                                                   float* __restrict__ vposed) {
    __shared__ float Atile[BM_ * AST_];   // 32 x 488 floats, TDM-filled
    __shared__ float Btas[BM_ * NB_];     // betas tile
    const int tid = threadIdx.x;
    const int m0 = blockIdx.y * BM_;
    const int n0 = blockIdx.x * BN_;
    const int wave = tid >> 5, lane = tid & 31;

    // --- TDM: each wave DMAs its own 4-row slice of the pose-feature tile ---
    {
        const int r0 = wave * 4;  // rows [r0, r0+4) of the 32-row tile
        unsigned long long ga =
            (unsigned long long)(size_t)(pf_buf + (size_t)(m0 + r0) * KP_);
        unsigned lds_off =
            (unsigned)(size_t)(&Atile[r0 * AST_]);   // addrspace(3) byte offset
        u32x4 g0;
        g0[0] = 1u;                                   // count=1, no gather
        g0[1] = lds_off;                              // lds_addr
        g0[2] = (unsigned)ga;                         // global_addr[31:0]
        g0[3] = ((unsigned)(ga >> 32) & 0x01FFFFFFu) | 0x80000000u; // addr[56:32]|type=2
        i32x8 g1;
        g1[0] = (int)(2u << 16);        // data_size = 4 bytes
        g1[1] = (int)(488u << 16);      // tensor_dim0 low16 (bits 63:48)
        g1[2] = (int)(4u << 16);        // tensor_dim0 hi=0 | tensor_dim1=4 (bits 95:80)
        g1[3] = (int)(488u << 16);      // tensor_dim1 hi=0 | tile_dim0=488 (127:112)
        g1[4] = 4;                      // tile_dim1=4, tile_dim2=0
        g1[5] = 488;                    // tensor_dim0_stride low32
        g1[6] = 0;                      // stride hi / dim1_stride
        g1[7] = 0;
        i32x4 gz4 = {0, 0, 0, 0};
        i32x8 gz8 = {0, 0, 0, 0, 0, 0, 0, 0};
        __builtin_amdgcn_tensor_load_to_lds(g0, g1, gz4, gz4, gz8, 0);
        __builtin_amdgcn_s_wait_tensorcnt(0);
    }
    for (int i = tid; i < BM_ * NB_; i += 256)
        Btas[i] = betas[(size_t)m0 * NB_ + i];
    __syncthreads();

    const int half = lane >> 4, nl = lane & 15;
    const int n = n0 + wave * 16 + nl;      // output column (v*3+c)
    const bool nok = (n < NCOLS);
    const int ncl = nok ? n : (NCOLS - 1);  // clamped: dead lanes read valid addr,
                                            // garbage lands only in unstored cols
    const float* pd = posedirs + ncl;
    const int ko = half * 2;                // lanes 16-31 own K pair {k+2,k+3}

    v8f acc0 = {}; v8f acc1 = {};
    #pragma unroll 2
    for (int k = 0; k < 484; k += 4) {      // kp max = 483 < 486: no guards
        int kp = k + ko;
        v2f a0, a1, b;
        a0.x = Atile[nl * AST_ + kp];        a0.y = Atile[nl * AST_ + kp + 1];
        a1.x = Atile[(nl + 16) * AST_ + kp]; a1.y = Atile[(nl + 16) * AST_ + kp + 1];
        b.x = pd[(size_t)kp * NCOLS];
        b.y = pd[(size_t)(kp + 1) * NCOLS];
        acc0 = __builtin_amdgcn_wmma_f32_16x16x4_f32(false, a0, false, b,
                                                     (short)0, acc0, false, false);
        acc1 = __builtin_amdgcn_wmma_f32_16x16x4_f32(false, a1, false, b,
                                                     (short)0, acc1, false, false);
    }
    {   // tail k=484: lanes with half=1 hit K rows 486/487 -> zero B
        int kp = 484 + ko;
        v2f a0, a1, b;
        a0.x = Atile[nl * AST_ + kp];        a0.y = Atile[nl * AST_ + kp + 1];
        a1.x = Atile[(nl + 16) * AST_ + kp]; a1.y = Atile[(nl + 16) * AST_ + kp + 1];
        float bx = 0.f, by = 0.f;
        if (kp < PF_)     bx = pd[(size_t)kp * NCOLS];
        if (kp + 1 < PF_) by = pd[(size_t)(kp + 1) * NCOLS];
        b.x = bx; b.y = by;
        acc0 = __builtin_amdgcn_wmma_f32_16x16x4_f32(false, a0, false, b,
                                                     (short)0, acc0, false, false);
        acc1 = __builtin_amdgcn_wmma_f32_16x16x4_f32(false, a1, false, b,
                                                     (short)0, acc1, false, false);
    }
    if (!nok) return;

    // fused epilogue: + v_template + betas . shapedirs
    float vt = v_template[n];
    float sd[NB_];
    #pragma unroll
    for (int k = 0; k < NB_; ++k) sd[k] = shapedirs[(size_t)n * NB_ + k];
    #pragma unroll
    for (int r = 0; r < 8; ++r) {
        int ml = half * 8 + r;   // local row within 16-row subtile
        float s0 = vt, s1 = vt;
        #pragma unroll
        for (int k = 0; k < NB_; ++k) {
            s0 += Btas[ml * NB_ + k] * sd[k];
            s1 += Btas[(ml + 16) * NB_ + k] * sd[k];
        }
        vposed[(size_t)(m0 + ml) * NCOLS + n]      = acc0[r] + s0;
        vposed[(size_t)(m0 + 16 + ml) * NCOLS + n] = acc1[r] + s1;
    }
}

// ---------------------------------------------------------------------------
// Kernel 3: skinning.  verts = (sum_j w[v][j] A[f][j])[:3,:4] applied to v_posed
// ---------------------------------------------------------------------------
__global__ __launch_bounds__(256) void k_skin(const float* __restrict__ wT,
                                              const float* __restrict__ A_buf,
                                              const float* __restrict__ vposed,
                                              const float* __restrict__ transl,
                                              float* __restrict__ out) {
    const int f = blockIdx.y;
    const int v = blockIdx.x * 256 + threadIdx.x;
    __shared__ float As[J_ * 16];
    __shared__ float tr[3];
    for (int i = threadIdx.x; i < J_ * 16; i += 256)
        As[i] = A_buf[(size_t)f * (J_ * 16) + i];
    if (threadIdx.x < 3) tr[threadIdx.x] = transl[(size_t)f * 3 + threadIdx.x];
    __syncthreads();
    if (v >= V_) return;

    float T[12];
    #pragma unroll
    for (int e = 0; e < 12; ++e) T[e] = 0.f;
    for (int j = 0; j < J_; ++j) {
        float w = wT[(size_t)j * V_ + v];   // coalesced, L2-resident
        #pragma unroll
        for (int e = 0; e < 12; ++e) T[e] += w * As[j * 16 + e];
    }
    const float* vp = vposed + (size_t)f * NCOLS + (size_t)v * 3;
    float p0 = vp[0], p1 = vp[1], p2 = vp[2];
    float* o = out + ((size_t)f * V_ + v) * 3;
    o[0] = T[0] * p0 + T[1] * p1 + T[2]  * p2 + T[3]  + tr[0];
    o[1] = T[4] * p0 + T[5] * p1 + T[6]  * p2 + T[7]  + tr[1];
    o[2] = T[8] * p0 + T[9] * p1 + T[10] * p2 + T[11] + tr[2];
}

// ---------------------------------------------------------------------------
extern "C" void kernel_launch(void* const* d_in, const int* in_sizes, int n_in,
                              void* d_out, int out_size, void* d_ws, size_t ws_size,
                              hipStream_t stream) {
    const float* body_pose     = (const float*)d_in[0];
    const float* betas         = (const float*)d_in[1];
    const float* global_orient = (const float*)d_in[2];
    const float* transl        = (const float*)d_in[3];
    const float* v_template    = (const float*)d_in[4];
    const float* shapedirs     = (const float*)d_in[5];
    const float* posedirs      = (const float*)d_in[6];
    const float* lbs_weights   = (const float*)d_in[7];
    const float* J_template    = (const float*)d_in[8];
    const float* J_shapedirs   = (const float*)d_in[9];
    const float* odp           = (const float*)d_in[10];
    float* out = (float*)d_out;

    float* ws = (float*)d_ws;
    float* pf_buf = ws;                                   // 512*488
    float* A_buf  = pf_buf + (size_t)FR_ * KP_;           // 512*880
    float* wT     = A_buf + (size_t)FR_ * J_ * 16;        // 55*10475
    float* vposed = ws + 1276544;                         // 512*31425 (16-aligned)

    k_transpose_w<<<(V_ * J_ + 255) / 256, 256, 0, stream>>>(lbs_weights, wT);
    k_pose<<<FR_, 64, 0, stream>>>(body_pose, betas, global_orient,
                                   J_template, J_shapedirs, odp, pf_buf, A_buf);
    dim3 ggrid((NCOLS + BN_ - 1) / BN_, FR_ / BM_);       // (246, 16)
    k_gemm_pose<<<ggrid, 256, 0, stream>>>(pf_buf, posedirs, shapedirs,
                                           v_template, betas, vposed);
    dim3 sgrid((V_ + 255) / 256, FR_);                    // (41, 512)
    k_skin<<<sgrid, 256, 0, stream>>>(wT, A_buf, vposed, transl, out);
}